// DynamicGNN_6914897347005
// MI455X (gfx1250) — compile-verified
//
#include <hip/hip_runtime.h>
#include <hip/hip_bf16.h>
#include <math.h>

#define N_NODES   10000
#define N_EDGES   320000
#define HID       256
#define HEADS     8
#define HEAD_DIM  32
#define N_LAYERS  3
#define LEAKY_A   0.2f
#define LN_EPS    1e-5f

typedef __attribute__((ext_vector_type(2))) float v2f;
typedef __attribute__((ext_vector_type(8))) float v8f;

// ---------------------------------------------------------------------------
// GEMM: h = x @ W^T + b   using V_WMMA_F32_16X16X4_F32 (full fp32 matrix op).
// Grid: 625 row-tiles (10000/16 exact). Block = 128 thr = 4 waves.
// Each wave owns 4 column tiles (16 col-tiles total); A loaded once per
// k-step and reused across the 4 WMMAs.
// A[m][k] = x[row0+m][kb+k] ; B[k][n] = W[col0+n][kb+k]  (both float2/lane)
// ---------------------------------------------------------------------------
__global__ __launch_bounds__(128)
void gat_gemm_wmma(const float* __restrict__ x, const float* __restrict__ W,
                   const float* __restrict__ bias, float* __restrict__ h)
{
    const int wave   = threadIdx.x >> 5;   // 0..3
    const int lane   = threadIdx.x & 31;
    const int laneLo = lane & 15;
    const int hi     = lane >> 4;          // 0 or 1
    const int row0   = blockIdx.x * 16;

    const float* arow = x + (size_t)(row0 + laneLo) * HID + 2 * hi;
    // 4 column tiles per wave: ct = wave*4 + t
    const float* brow[4];
    int col0[4];
#pragma unroll
    for (int t = 0; t < 4; ++t) {
        col0[t] = (wave * 4 + t) * 16;
        brow[t] = W + (size_t)(col0[t] + laneLo) * HID + 2 * hi;
    }

    v8f acc[4] = {v8f{}, v8f{}, v8f{}, v8f{}};

#pragma unroll 4
    for (int k = 0; k < HID; k += 4) {
        v2f a = *(const v2f*)(arow + k);
#pragma unroll
        for (int t = 0; t < 4; ++t) {
            v2f b = *(const v2f*)(brow[t] + k);
            acc[t] = __builtin_amdgcn_wmma_f32_16x16x4_f32(
                false, a, false, b, (short)0, acc[t], false, false);
        }
    }

#pragma unroll
    for (int t = 0; t < 4; ++t) {
        const float bv = bias[col0[t] + laneLo];   // N = laneLo
#pragma unroll
        for (int v = 0; v < 8; ++v) {
            const int m = v + 8 * hi;              // C layout: M = vgpr + 8*hiHalf
            h[(size_t)(row0 + m) * HID + col0[t] + laneLo] = acc[t][v] + bv;
        }
    }
}

// ---------------------------------------------------------------------------
// Per-node attention scores: s_src[n,h] = h[n,h,:]·a_src[h], same for a_dst.
// One thread per (node, head).
// ---------------------------------------------------------------------------
__global__ void gat_scores(const float* __restrict__ h,
                           const float* __restrict__ a_src,
                           const float* __restrict__ a_dst,
                           float* __restrict__ ssrc, float* __restrict__ sdst)
{
    int idx = blockIdx.x * blockDim.x + threadIdx.x;
    if (idx >= N_NODES * HEADS) return;
    const int n  = idx >> 3;
    const int hh = idx & 7;
    const float* hp = h + (size_t)n * HID + hh * HEAD_DIM;
    const float* as = a_src + hh * HEAD_DIM;
    const float* ad = a_dst + hh * HEAD_DIM;
    float s1 = 0.f, s2 = 0.f;
#pragma unroll
    for (int c = 0; c < HEAD_DIM; ++c) {
        const float v = hp[c];
        s1 = fmaf(v, as[c], s1);
        s2 = fmaf(v, ad[c], s2);
    }
    ssrc[idx] = s1;
    sdst[idx] = s2;
}

// ---------------------------------------------------------------------------
// Init per-layer accumulators: amax=-inf, denom=0, agg=0.
// ---------------------------------------------------------------------------
__global__ void gat_init(float* __restrict__ amax, float* __restrict__ denom,
                         float* __restrict__ agg)
{
    int idx = blockIdx.x * blockDim.x + threadIdx.x;
    if (idx < N_NODES * HID) agg[idx] = 0.f;
    if (idx < N_NODES * HEADS) {
        amax[idx]  = -INFINITY;
        denom[idx] = 0.f;
    }
}

// Float atomic max via IEEE bit ordering (lowers to native int atomics).
__device__ __forceinline__ void atomicMaxF(float* addr, float val)
{
    if (val >= 0.f)
        atomicMax((int*)addr, __float_as_int(val));
    else
        atomicMin((unsigned int*)addr, __float_as_uint(val));
}

// ---------------------------------------------------------------------------
// Pass A: alpha[e,h] = leaky(ssrc[dst,h] + sdst[src,h]); atomic-max into amax.
// ---------------------------------------------------------------------------
__global__ void gat_edge_alpha(const long long* __restrict__ esrc,
                               const long long* __restrict__ edst,
                               const float* __restrict__ ssrc,
                               const float* __restrict__ sdst,
                               float* __restrict__ ebuf,
                               float* __restrict__ amax)
{
    int idx = blockIdx.x * blockDim.x + threadIdx.x;
    if (idx >= N_EDGES * HEADS) return;
    const int e  = idx >> 3;
    const int hh = idx & 7;
    const int s  = (int)esrc[e];
    const int d  = (int)edst[e];
    float a = ssrc[d * HEADS + hh] + sdst[s * HEADS + hh];
    a = (a > 0.f) ? a : LEAKY_A * a;
    ebuf[idx] = a;
    atomicMaxF(&amax[d * HEADS + hh], a);
}

// ---------------------------------------------------------------------------
// Pass B: e = exp(alpha - amax[dst]); atomic-add into denom; store e.
// ---------------------------------------------------------------------------
__global__ void gat_edge_exp(const long long* __restrict__ edst,
                             float* __restrict__ ebuf,
                             const float* __restrict__ amax,
                             float* __restrict__ denom)
{
    int idx = blockIdx.x * blockDim.x + threadIdx.x;
    if (idx >= N_EDGES * HEADS) return;
    const int e  = idx >> 3;
    const int hh = idx & 7;
    const int d  = (int)edst[e];
    const float ev = __expf(ebuf[idx] - amax[d * HEADS + hh]);
    ebuf[idx] = ev;
    atomicAdd(&denom[d * HEADS + hh], ev);
}

// ---------------------------------------------------------------------------
// Pass C (bandwidth-dominant): one wave32 per edge.
// Lane l handles head l>>2, channels [head*32 + (l&3)*8 , +8): two float4
// loads of h[src], 8 GLOBAL_ATOMIC_ADD_F32 into agg[dst].
// ---------------------------------------------------------------------------
__global__ void gat_edge_msg(const long long* __restrict__ esrc,
                             const long long* __restrict__ edst,
                             const float* __restrict__ h,
                             const float* __restrict__ ebuf,
                             const float* __restrict__ denom,
                             float* __restrict__ agg)
{
    const int warpId = (blockIdx.x * blockDim.x + threadIdx.x) >> 5;
    const int lane   = threadIdx.x & 31;
    if (warpId >= N_EDGES) return;
    const int s    = (int)esrc[warpId];
    const int d    = (int)edst[warpId];
    const int head = lane >> 2;
    const float attn = ebuf[warpId * HEADS + head] /
                       (denom[d * HEADS + head] + 1e-16f);
    const int cbase = head * HEAD_DIM + (lane & 3) * 8;
    const float4 v0 = *(const float4*)(h + (size_t)s * HID + cbase);
    const float4 v1 = *(const float4*)(h + (size_t)s * HID + cbase + 4);
    float* op = agg + (size_t)d * HID + cbase;
    atomicAdd(op + 0, v0.x * attn);
    atomicAdd(op + 1, v0.y * attn);
    atomicAdd(op + 2, v0.z * attn);
    atomicAdd(op + 3, v0.w * attn);
    atomicAdd(op + 4, v1.x * attn);
    atomicAdd(op + 5, v1.y * attn);
    atomicAdd(op + 6, v1.z * attn);
    atomicAdd(op + 7, v1.w * attn);
}

// ---------------------------------------------------------------------------
// Fused (optional ReLU) + LayerNorm + residual. One wave32 per node,
// 8 channels per lane, wave32 shuffle reductions.
// ---------------------------------------------------------------------------
__global__ void gat_ln_res(const float* __restrict__ agg,
                           const float* __restrict__ xin,
                           const float* __restrict__ g,
                           const float* __restrict__ b,
                           float* __restrict__ xout, int do_relu)
{
    const int n    = (blockIdx.x * blockDim.x + threadIdx.x) >> 5;
    const int lane = threadIdx.x & 31;
    if (n >= N_NODES) return;

    float vals[8];
    float sum = 0.f;
#pragma unroll
    for (int j = 0; j < 8; ++j) {
        float v = agg[(size_t)n * HID + lane + j * 32];
        if (do_relu) v = fmaxf(v, 0.f);
        vals[j] = v;
        sum += v;
    }
#pragma unroll
    for (int m = 16; m >= 1; m >>= 1) sum += __shfl_xor(sum, m, 32);
    const float mean = sum * (1.f / HID);

    float var = 0.f;
#pragma unroll
    for (int j = 0; j < 8; ++j) {
        const float dl = vals[j] - mean;
        var = fmaf(dl, dl, var);
    }
#pragma unroll
    for (int m = 16; m >= 1; m >>= 1) var += __shfl_xor(var, m, 32);
    const float rstd = rsqrtf(var * (1.f / HID) + LN_EPS);

#pragma unroll
    for (int j = 0; j < 8; ++j) {
        const int c = lane + j * 32;
        const float nv = (vals[j] - mean) * rstd * g[c] + b[c];
        xout[(size_t)n * HID + c] = xin[(size_t)n * HID + c] + nv;
    }
}

// ---------------------------------------------------------------------------
extern "C" void kernel_launch(void* const* d_in, const int* in_sizes, int n_in,
                              void* d_out, int out_size, void* d_ws, size_t ws_size,
                              hipStream_t stream)
{
    const float*     x     = (const float*)d_in[0];
    const long long* eidx  = (const long long*)d_in[1];   // int64 per reference
    const float*     W     = (const float*)d_in[2];
    const float*     B     = (const float*)d_in[3];
    const float*     Asrc  = (const float*)d_in[4];
    const float*     Adst  = (const float*)d_in[5];
    const float*     LNg   = (const float*)d_in[6];
    const float*     LNb   = (const float*)d_in[7];

    const long long* esrc = eidx;
    const long long* edst = eidx + N_EDGES;

    // Workspace carve-up (aligned to 256B)
    char* p = (char*)d_ws;
    auto carve = [&](size_t bytes) {
        char* r = p;
        p += (bytes + 255) & ~(size_t)255;
        return (float*)r;
    };
    float* xbuf  = carve((size_t)N_NODES * HID * 4);     // running features
    float* hbuf  = carve((size_t)N_NODES * HID * 4);     // transformed features
    float* agg   = carve((size_t)N_NODES * HID * 4);     // aggregated messages
    float* ebuf  = carve((size_t)N_EDGES * HEADS * 4);   // alpha -> exp(alpha)
    float* ssrc  = carve((size_t)N_NODES * HEADS * 4);
    float* sdst  = carve((size_t)N_NODES * HEADS * 4);
    float* amax  = carve((size_t)N_NODES * HEADS * 4);
    float* denom = carve((size_t)N_NODES * HEADS * 4);

    hipMemcpyAsync(xbuf, x, (size_t)N_NODES * HID * 4,
                   hipMemcpyDeviceToDevice, stream);

    const int TPB = 256;
    const int gScore = (N_NODES * HEADS + TPB - 1) / TPB;
    const int gInit  = (N_NODES * HID + TPB - 1) / TPB;
    const int gEdgeH = (N_EDGES * HEADS + TPB - 1) / TPB;
    const int gMsg   = (N_EDGES * 32 + TPB - 1) / TPB;
    const int gLN    = (N_NODES * 32 + TPB - 1) / TPB;

    for (int l = 0; l < N_LAYERS; ++l) {
        const float* Wl  = W    + (size_t)l * HID * HID;
        const float* Bl  = B    + (size_t)l * HID;
        const float* ASl = Asrc + (size_t)l * HEADS * HEAD_DIM;
        const float* ADl = Adst + (size_t)l * HEADS * HEAD_DIM;
        const float* gl  = LNg  + (size_t)l * HID;
        const float* bl  = LNb  + (size_t)l * HID;

        gat_gemm_wmma<<<N_NODES / 16, 128, 0, stream>>>(xbuf, Wl, Bl, hbuf);
        gat_scores<<<gScore, TPB, 0, stream>>>(hbuf, ASl, ADl, ssrc, sdst);
        gat_init<<<gInit, TPB, 0, stream>>>(amax, denom, agg);
        gat_edge_alpha<<<gEdgeH, TPB, 0, stream>>>(esrc, edst, ssrc, sdst,
                                                   ebuf, amax);
        gat_edge_exp<<<gEdgeH, TPB, 0, stream>>>(edst, ebuf, amax, denom);
        gat_edge_msg<<<gMsg, TPB, 0, stream>>>(esrc, edst, hbuf, ebuf,
                                               denom, agg);
        float* xout = (l == N_LAYERS - 1) ? (float*)d_out : xbuf;
        gat_ln_res<<<gLN, TPB, 0, stream>>>(agg, xbuf, gl, bl, xout,
                                            (l < N_LAYERS - 1) ? 1 : 0);
    }
}